// Critic_73916387164188
// MI455X (gfx1250) — compile-verified
//
#include <hip/hip_runtime.h>
#include <math.h>

#define BB   256   // batch
#define NN   256   // nodes
#define FF   67    // features
#define HH   400   // hidden

typedef __attribute__((ext_vector_type(2))) float v2f;
typedef __attribute__((ext_vector_type(8))) float v8f;

__device__ __forceinline__ float wred_sum(float v) {
#pragma unroll
  for (int o = 16; o > 0; o >>= 1) v += __shfl_xor(v, o, 32);
  return v;
}
__device__ __forceinline__ float wred_max(float v) {
#pragma unroll
  for (int o = 16; o > 0; o >>= 1) v = fmaxf(v, __shfl_xor(v, o, 32));
  return v;
}

// ---- 1) w-vectors: wv[v] = W @ a_half  (4 vectors of length 67) ----
__global__ void k_wvec(const float* __restrict__ u1W, const float* __restrict__ u1a,
                       const float* __restrict__ d1W, const float* __restrict__ d1a,
                       float* __restrict__ wv) {
  int v = blockIdx.x;  // 0:u,a1  1:u,a2  2:d,a1  3:d,a2
  const float* W  = (v < 2) ? u1W : d1W;
  const float* av = ((v < 2) ? u1a : d1a) + ((v & 1) ? HH : 0);
  for (int f = threadIdx.x; f < FF; f += blockDim.x) {
    float s = 0.f;
    for (int h = 0; h < HH; ++h) s += W[f * HH + h] * av[h];
    wv[v * FF + f] = s;
  }
}

// ---- 2) ego = [x | a | 0 0]  (256 x 64, K padded) ----
__global__ void k_ego(const float* __restrict__ x, const float* __restrict__ a,
                      float* __restrict__ ego) {
  int b = blockIdx.x, t = threadIdx.x;  // block 64
  float v = 0.f;
  if (t < 61)       v = x[b * 61 + t];
  else if (t == 61) v = a[b];
  ego[b * 64 + t] = v;
}

// ---- 2b) zero-padded weight copy: Wp[Kpad x 400] = W[Kreal x 400] (pad rows = 0) ----
__global__ void k_padW(const float* __restrict__ W, float* __restrict__ Wp,
                       int Kreal, int total /* = Kpad*400 */) {
  int i = blockIdx.x * 256 + threadIdx.x;
  if (i < total) {
    int row = i / HH;
    Wp[i] = (row < Kreal) ? W[i] : 0.f;
  }
}

// ---- 3) attention scores: per (b,j) row, 4 dot products with wv ----
__global__ void k_scores(const float* __restrict__ fp, const float* __restrict__ wv,
                         float* __restrict__ s_u2, float* __restrict__ s_d2,
                         float* __restrict__ s_u1, float* __restrict__ s_d1) {
  int lane = threadIdx.x & 31;
  int w    = threadIdx.x >> 5;
  int row  = blockIdx.x * 8 + w;            // < B*N
  const float* r   = fp + (size_t)row * FF;
  const float* wu1 = wv;
  const float* wu2 = wv + FF;
  const float* wd1 = wv + 2 * FF;
  const float* wd2 = wv + 3 * FF;
  float du1 = 0.f, du2 = 0.f, dd1 = 0.f, dd2 = 0.f;
  for (int f = lane; f < FF; f += 32) {
    float xv = r[f];
    du1 += xv * wu1[f]; du2 += xv * wu2[f];
    dd1 += xv * wd1[f]; dd2 += xv * wd2[f];
  }
  du1 = wred_sum(du1); du2 = wred_sum(du2);
  dd1 = wred_sum(dd1); dd2 = wred_sum(dd2);
  if (lane == 0) {
    s_u2[row] = du2; s_d2[row] = dd2;
    if ((row & (NN - 1)) == 0) { int b = row >> 8; s_u1[b] = du1; s_d1[b] = dd1; }
  }
}

// ---- 4) softmax over j=1..255 of leaky_relu(s1[b] + s2[b,j]); att[b,0] = 0 ----
__global__ void k_softmax(const float* __restrict__ s1, const float* __restrict__ s2,
                          float* __restrict__ att) {
  __shared__ float smax[8];
  __shared__ float ssum[8];
  int b = blockIdx.x, j = threadIdx.x;      // block 256
  int lane = j & 31, w = j >> 5;
  float e;
  if (j == 0) e = -3.0e38f;
  else {
    float t = s1[b] + s2[b * NN + j];
    e = (t > 0.f) ? t : 0.2f * t;           // leaky_relu alpha=0.2
  }
  float m = wred_max(e);
  if (lane == 0) smax[w] = m;
  __syncthreads();
  m = smax[0];
#pragma unroll
  for (int q = 1; q < 8; ++q) m = fmaxf(m, smax[q]);
  float ex = (j == 0) ? 0.f : expf(e - m);
  float s = wred_sum(ex);
  if (lane == 0) ssum[w] = s;
  __syncthreads();
  float tot = 0.f;
#pragma unroll
  for (int q = 0; q < 8; ++q) tot += ssum[q];
  att[b * NN + j] = ex / tot;
}

// ---- 5) fbar[b,f] = sum_j att[b,j]*fp[b,j,f]; also pack feat0 (stride 68, zero pad) ----
__global__ void k_fbar(const float* __restrict__ fp,
                       const float* __restrict__ att_u, const float* __restrict__ att_d,
                       float* __restrict__ fbar_u, float* __restrict__ fbar_d,
                       float* __restrict__ feat0) {
  int b = blockIdx.x, f = threadIdx.x;      // block 96
  float au = 0.f, ad = 0.f;
  if (f < FF) {
    const float* base = fp + (size_t)b * NN * FF + f;
    const float* atu  = att_u + b * NN;
    const float* atd  = att_d + b * NN;
    for (int j = 1; j < NN; ++j) {
      float v = base[j * FF];
      au += atu[j] * v;
      ad += atd[j] * v;
    }
  }
  if (f < 68) {
    fbar_u[b * 68 + f] = (f < FF) ? au : 0.f;
    fbar_d[b * 68 + f] = (f < FF) ? ad : 0.f;
    feat0[b * 68 + f]  = (f < FF) ? fp[(size_t)b * NN * FF + f] : 0.f;
  }
}

// ---- 6) fp32 WMMA GEMM: C[256 x 400] = A[256 x Kpad] @ Bw[Kpad x 400] (+bias)(+relu)
// A and Bw are zero-padded to Kpad -> NO bounds checks in the hot loop.
// One wave per block; wave computes 16 rows x 80 cols (5 accumulators, A reused).
__global__ void k_gemm400(const float* __restrict__ A, int lda,
                          const float* __restrict__ Bw,
                          float* __restrict__ C,
                          int Kpad,
                          const float* __restrict__ bias, int act) {
  const int N = HH;                          // 400
  int lane = threadIdx.x;                    // 0..31
  int l16  = lane & 15;
  int half = lane >> 4;                      // 0 or 1
  int tm   = blockIdx.y * 16;
  int tn0  = blockIdx.x * 80;
  v8f acc[5] = {};
  const float* arow = A + (size_t)(tm + l16) * lda;
  const float* bcol = Bw + tn0 + l16;
  for (int k = 0; k < Kpad; k += 4) {
    int ka = k + half * 2;                   // A: lane l16 holds row m, K = ka, ka+1
    v2f av = *(const v2f*)(arow + ka);       // 8B aligned: lda*4 % 16 == 0, ka even
    const float* b0 = bcol + (size_t)ka * N;
#pragma unroll
    for (int t = 0; t < 5; ++t) {
      v2f bv;                                // B: lane l16 holds col n, K = ka, ka+1
      bv.x = b0[t * 16];
      bv.y = b0[N + t * 16];
      acc[t] = __builtin_amdgcn_wmma_f32_16x16x4_f32(
          false, av, false, bv, (short)0, acc[t], false, false);
    }
  }
#pragma unroll
  for (int t = 0; t < 5; ++t) {
    int n = tn0 + t * 16 + l16;
    float bsv = bias ? bias[n] : 0.f;
#pragma unroll
    for (int r = 0; r < 8; ++r) {
      int m = tm + r + half * 8;             // C: VGPR r -> row r (lanes 0-15) / r+8
      float v = acc[t][r] + bsv;
      if (act) v = (v > 0.f) ? v : 0.f;
      C[(size_t)m * N + n] = v;
    }
  }
}

// ---- 7) z = elu(Wh0) + 255*elu(hp0)   (layer-2 input collapsed by star topology) ----
__global__ void k_z(const float* __restrict__ Wh0, const float* __restrict__ hp0,
                    float* __restrict__ z) {
  int i = blockIdx.x * 256 + threadIdx.x;   // 102400 total
  float w = Wh0[i], p = hp0[i];
  float ew = (w > 0.f) ? w : expm1f(w);
  float ep = (p > 0.f) ? p : expm1f(p);
  z[i] = ew + 255.f * ep;
}

// ---- 8) final heads: A=(S_u+S_d)@fc3_w+b3, Q=h2@fc2_w+b2, G=Q+A, reg=0 ----
__global__ void k_final(const float* __restrict__ S_u, const float* __restrict__ S_d,
                        const float* __restrict__ h2,
                        const float* __restrict__ fc3_w, const float* __restrict__ fc3_b,
                        const float* __restrict__ fc2_w, const float* __restrict__ fc2_b,
                        float* __restrict__ out) {
  __shared__ float sA[8];
  __shared__ float sQ[8];
  int b = blockIdx.x, t = threadIdx.x;      // block 256
  int lane = t & 31, w = t >> 5;
  float accA = 0.f, accQ = 0.f;
  for (int h = t; h < HH; h += 256) {
    accA += (S_u[b * HH + h] + S_d[b * HH + h]) * fc3_w[h];
    accQ += h2[b * HH + h] * fc2_w[h];
  }
  accA = wred_sum(accA); accQ = wred_sum(accQ);
  if (lane == 0) { sA[w] = accA; sQ[w] = accQ; }
  __syncthreads();
  if (t == 0) {
    float Av = fc3_b[0], Qv = fc2_b[0];
#pragma unroll
    for (int q = 0; q < 8; ++q) { Av += sA[q]; Qv += sQ[q]; }
    out[b]       = Qv;          // Q
    out[256 + b] = Av;          // A
    out[512 + b] = Qv + Av;     // G
    if (b == 0) out[768] = 0.f; // reg
  }
}

extern "C" void kernel_launch(void* const* d_in, const int* in_sizes, int n_in,
                              void* d_out, int out_size, void* d_ws, size_t ws_size,
                              hipStream_t stream) {
  (void)in_sizes; (void)n_in; (void)out_size; (void)ws_size;
  const float* x     = (const float*)d_in[0];
  const float* a     = (const float*)d_in[1];
  const float* fp    = (const float*)d_in[2];
  // d_in[3] embed: unused by reference
  const float* fc0_w = (const float*)d_in[4];
  const float* fc0_b = (const float*)d_in[5];
  const float* fc1_w = (const float*)d_in[6];
  const float* fc1_b = (const float*)d_in[7];
  const float* fc2_w = (const float*)d_in[8];
  const float* fc2_b = (const float*)d_in[9];
  const float* fc3_w = (const float*)d_in[10];
  const float* fc3_b = (const float*)d_in[11];
  const float* u1_W  = (const float*)d_in[12];
  const float* u1_a  = (const float*)d_in[13];
  const float* u2_W  = (const float*)d_in[14];
  // u2_a (15): provably does not affect output (uniform softmax)
  const float* d1_W  = (const float*)d_in[16];
  const float* d1_a  = (const float*)d_in[17];
  const float* d2_W  = (const float*)d_in[18];
  // d2_a (19): provably does not affect output
  float* out = (float*)d_out;

  float* ws = (float*)d_ws;
  size_t off = 0;
  auto alloc = [&](size_t n) { float* p = ws + off; off += n; return p; };
  float* wv     = alloc(320);             // 4 x 67
  float* s_u2   = alloc(BB * NN);
  float* s_d2   = alloc(BB * NN);
  float* s_u1   = alloc(BB);
  float* s_d1   = alloc(BB);
  float* att_u  = alloc(BB * NN);
  float* att_d  = alloc(BB * NN);
  float* feat0  = alloc(BB * 68);
  float* fbar_u = alloc(BB * 68);
  float* fbar_d = alloc(BB * 68);
  float* ego    = alloc(BB * 64);
  float* u1_Wp  = alloc(68 * HH);         // zero-padded weights (no guards in GEMM)
  float* d1_Wp  = alloc(68 * HH);
  float* fc0_wp = alloc(64 * HH);
  float* Whu0   = alloc(BB * HH);
  float* hpu0   = alloc(BB * HH);
  float* Whd0   = alloc(BB * HH);
  float* hpd0   = alloc(BB * HH);
  float* z_u    = alloc(BB * HH);
  float* z_d    = alloc(BB * HH);
  float* S_u    = alloc(BB * HH);
  float* S_d    = alloc(BB * HH);
  float* h1     = alloc(BB * HH);
  float* h2     = alloc(BB * HH);

  k_wvec<<<4, 128, 0, stream>>>(u1_W, u1_a, d1_W, d1_a, wv);
  k_ego<<<BB, 64, 0, stream>>>(x, a, ego);
  k_padW<<<(68 * HH + 255) / 256, 256, 0, stream>>>(u1_W, u1_Wp, FF, 68 * HH);
  k_padW<<<(68 * HH + 255) / 256, 256, 0, stream>>>(d1_W, d1_Wp, FF, 68 * HH);
  k_padW<<<(64 * HH + 255) / 256, 256, 0, stream>>>(fc0_w, fc0_wp, 62, 64 * HH);
  k_scores<<<(BB * NN) / 8, 256, 0, stream>>>(fp, wv, s_u2, s_d2, s_u1, s_d1);
  k_softmax<<<BB, 256, 0, stream>>>(s_u1, s_u2, att_u);
  k_softmax<<<BB, 256, 0, stream>>>(s_d1, s_d2, att_d);
  k_fbar<<<BB, 96, 0, stream>>>(fp, att_u, att_d, fbar_u, fbar_d, feat0);

  dim3 ggrid(5, BB / 16);
  // layer-1 GATs (K padded to 68)
  k_gemm400<<<ggrid, 32, 0, stream>>>(feat0,  68, u1_Wp, Whu0, 68, nullptr, 0);
  k_gemm400<<<ggrid, 32, 0, stream>>>(fbar_u, 68, u1_Wp, hpu0, 68, nullptr, 0);
  k_gemm400<<<ggrid, 32, 0, stream>>>(feat0,  68, d1_Wp, Whd0, 68, nullptr, 0);
  k_gemm400<<<ggrid, 32, 0, stream>>>(fbar_d, 68, d1_Wp, hpd0, 68, nullptr, 0);
  // MLP layer 0 (K padded to 64)
  k_gemm400<<<ggrid, 32, 0, stream>>>(ego, 64, fc0_wp, h1, 64, fc0_b, 1);

  k_z<<<(BB * HH) / 256, 256, 0, stream>>>(Whu0, hpu0, z_u);
  k_z<<<(BB * HH) / 256, 256, 0, stream>>>(Whd0, hpd0, z_d);

  // layer-2 GAT node-sums + MLP layer 1 (K = 400)
  k_gemm400<<<ggrid, 32, 0, stream>>>(z_u, HH, u2_W, S_u, HH, nullptr, 0);
  k_gemm400<<<ggrid, 32, 0, stream>>>(z_d, HH, d2_W, S_d, HH, nullptr, 0);
  k_gemm400<<<ggrid, 32, 0, stream>>>(h1, HH, fc1_w, h2, HH, fc1_b, 1);

  k_final<<<BB, 256, 0, stream>>>(S_u, S_d, h2, fc3_w, fc3_b, fc2_w, fc2_b, out);
}